// TDrumorGCN_78847009620360
// MI455X (gfx1250) — compile-verified
//
#include <hip/hip_runtime.h>
#include <hip/hip_bf16.h>

// Dims fixed by the reference
#define IN_F  256
#define HID_F 128
#define OUT_F 128

typedef float v2f __attribute__((ext_vector_type(2)));
typedef float v8f __attribute__((ext_vector_type(8)));

// ---------------- utility ----------------
__global__ void zero_kernel(float* __restrict__ p, long n) {
  long i = (long)blockIdx.x * blockDim.x + threadIdx.x;
  if (i < n) p[i] = 0.0f;
}

// deg[dst] += 1 per edge
__global__ void deg_kernel(const int* __restrict__ dst, float* __restrict__ deg, int E) {
  int e = blockIdx.x * blockDim.x + threadIdx.x;
  if (e < E) atomicAdd(&deg[dst[e]], 1.0f);
}

// rootidx[g] = first node of graph g (batch sorted ascending)
__global__ void root_kernel(const int* __restrict__ batch, int* __restrict__ rootidx, int N) {
  int i = blockIdx.x * blockDim.x + threadIdx.x;
  if (i >= N) return;
  int g = batch[i];
  if (i == 0 || batch[i - 1] != g) rootidx[g] = i;
}

__global__ void dinv_kernel(const float* __restrict__ deg, float* __restrict__ dinv, int N) {
  int i = blockIdx.x * blockDim.x + threadIdx.x;
  if (i < N) dinv[i] = rsqrtf(deg[i] + 1.0f);
}

// ---------------- WMMA GEMM: C[M x 128] = op(A[M x K]) @ B[K x 128] ----------------
// block = 256 threads = 8 waves; each wave owns one 16x16 output tile.
// MODE 0: plain.  MODE 1: relu(A) on load, += bpart[batch[row]][col] on store.
template <int MODE>
__global__ void gemm16_wmma(const float* __restrict__ A, const float* __restrict__ Bm,
                            float* __restrict__ C, int M, int K,
                            const float* __restrict__ bpart, const int* __restrict__ batch) {
  const int lane = threadIdx.x & 31;
  const int wave = threadIdx.x >> 5;
  const int m0 = blockIdx.x * 16;
  const int n0 = wave * 16;          // 8 waves * 16 = 128 output cols
  const int r  = lane & 15;
  const int hb = lane >> 4;          // 0/1 lane-half

  int ar = m0 + r; if (ar >= M) ar = M - 1;   // clamp for ragged tail (loads only)
  const float* arow = A + (size_t)ar * K;
  const float* bcol = Bm + n0 + r;

  v8f acc = {};
  for (int k = 0; k < K; k += 4) {
#if __has_builtin(__builtin_amdgcn_wmma_f32_16x16x4_f32)
    float2 av = *(const float2*)(arow + k + 2 * hb);      // A[row][k+2hb .. +1]
    if (MODE == 1) { av.x = fmaxf(av.x, 0.0f); av.y = fmaxf(av.y, 0.0f); }
    v2f a = {av.x, av.y};
    v2f b;
    b.x = bcol[(size_t)(k + 2 * hb) * 128];               // B[k+2hb  ][n0+r]
    b.y = bcol[(size_t)(k + 2 * hb + 1) * 128];           // B[k+2hb+1][n0+r]
    acc = __builtin_amdgcn_wmma_f32_16x16x4_f32(
        /*neg_a=*/false, a, /*neg_b=*/false, b,
        /*c_mod=*/(short)0, acc, /*reuse_a=*/false, /*reuse_b=*/false);
#else
    // scalar fallback (host pass / builtin unavailable) — same tile mapping
    for (int j = 0; j < 8; ++j) {
      int row = m0 + j + 8 * hb; if (row >= M) row = M - 1;
      for (int kk = 0; kk < 4; ++kk) {
        float av2 = A[(size_t)row * K + k + kk];
        if (MODE == 1) av2 = fmaxf(av2, 0.0f);
        acc[j] += av2 * Bm[(size_t)(k + kk) * 128 + n0 + r];
      }
    }
#endif
  }

  // C layout: c[j] -> C[m0 + j + 8*hb][n0 + r]
  float* crow = C + (size_t)(m0 + 8 * hb) * 128 + n0 + r;
#pragma unroll
  for (int j = 0; j < 8; ++j) {
    int row = m0 + j + 8 * hb;
    if (row < M) {
      float v = acc[j];
      if (MODE == 1) v += bpart[(size_t)batch[row] * 128 + n0 + r];
      crow[(size_t)j * 128] = v;
    }
  }
}

// ---------------- edge scatter: s[dst] += h[src] * dinv[src] ----------------
// one wave per edge, float4 per lane (128 feats = 32 lanes * 4)
__global__ void edge_agg_kernel(const int* __restrict__ src, const int* __restrict__ dst,
                                const float* __restrict__ dinv, const float* __restrict__ h,
                                float* __restrict__ s, int E) {
  int wid  = (int)(((long)blockIdx.x * blockDim.x + threadIdx.x) >> 5);
  int lane = threadIdx.x & 31;
  if (wid >= E) return;
  int sn = src[wid];
  int dn = dst[wid];
  float w = dinv[sn];
  float4 v = *(const float4*)(h + (size_t)sn * 128 + lane * 4);
  float* sp = s + (size_t)dn * 128 + lane * 4;
  atomicAdd(sp + 0, v.x * w);
  atomicAdd(sp + 1, v.y * w);
  atomicAdd(sp + 2, v.z * w);
  atomicAdd(sp + 3, v.w * w);
}

// conv finalize (in place over h): h := dinv[i]*(s + h*dinv[i]) + bias[f]
__global__ void finalize_kernel(float* __restrict__ h, const float* __restrict__ s,
                                const float* __restrict__ dinv, const float* __restrict__ bias,
                                long total) {
  long idx = (long)blockIdx.x * blockDim.x + threadIdx.x;
  if (idx >= total) return;
  int f = (int)(idx & 127);
  long i = idx >> 7;
  float di = dinv[i];
  h[idx] = di * fmaf(h[idx], di, s[idx]) + bias[f];
}

// rootc1[g][f] = c1out[rootidx[g]][f]
__global__ void root_gather_kernel(const float* __restrict__ c1, const int* __restrict__ rootidx,
                                   float* __restrict__ rootc1) {
  int g = blockIdx.x, f = threadIdx.x;
  rootc1[(size_t)g * 128 + f] = c1[(size_t)rootidx[g] * 128 + f];
}

// bpart[g][c] = relu(x[root_g]) @ W2[128:384, c]   (tiny per-graph GEMV)
__global__ void root_part_kernel(const float* __restrict__ x, const float* __restrict__ W2,
                                 const int* __restrict__ rootidx, float* __restrict__ bpart) {
  int g = blockIdx.x, c = threadIdx.x;
  const float* xr = x + (size_t)rootidx[g] * IN_F;
  const float* w  = W2 + (size_t)HID_F * 128 + c;   // rows 128..383 of W2 (384x128)
  float acc = 0.0f;
  for (int k = 0; k < IN_F; ++k)
    acc = fmaf(fmaxf(xr[k], 0.0f), w[(size_t)k * 128], acc);
  bpart[(size_t)g * 128 + c] = acc;
}

// conv2 finalize + relu + per-graph sum into out[:, 0:128].
// batch is sorted -> local run-accumulate, flush one atomic per graph change.
__global__ void finalize2_pool_kernel(const float* __restrict__ h2, const float* __restrict__ s2,
                                      const float* __restrict__ dinv, const float* __restrict__ b2,
                                      const int* __restrict__ batch, float* __restrict__ out,
                                      int N) {
  int f   = threadIdx.x & 127;
  int sub = threadIdx.x >> 7;           // 0/1
  long i0 = (long)blockIdx.x * 64;
  long iend = i0 + 64; if (iend > N) iend = N;
  float acc = 0.0f; int curg = -1;
  for (long i = i0 + sub; i < iend; i += 2) {
    int g = batch[i];
    if (g != curg) {
      if (curg >= 0) atomicAdd(&out[(size_t)curg * 256 + f], acc);
      curg = g; acc = 0.0f;
    }
    float di = dinv[i];
    long idx = i * 128 + f;
    float v = di * fmaf(h2[idx], di, s2[idx]) + b2[f];
    acc += fmaxf(v, 0.0f);
  }
  if (curg >= 0) atomicAdd(&out[(size_t)curg * 256 + f], acc);
}

// out[g][0:128] /= count_g ; out[g][128:256] = rootc1[g]  (counts from rootidx diffs)
__global__ void final_out_kernel(float* __restrict__ out, const float* __restrict__ rootc1,
                                 const int* __restrict__ rootidx, int N, int B) {
  int g = blockIdx.x, t = threadIdx.x;
  int start = rootidx[g];
  int end   = (g == B - 1) ? N : rootidx[g + 1];
  float cnt = (float)((end - start) > 1 ? (end - start) : 1);
  if (t < 128) out[(size_t)g * 256 + t] /= cnt;
  else         out[(size_t)g * 256 + t] = rootc1[(size_t)g * 128 + (t - 128)];
}

// ---------------- driver ----------------
extern "C" void kernel_launch(void* const* d_in, const int* in_sizes, int n_in,
                              void* d_out, int out_size, void* d_ws, size_t ws_size,
                              hipStream_t stream) {
  const float* x   = (const float*)d_in[0];
  const float* W1  = (const float*)d_in[1];
  const float* b1  = (const float*)d_in[2];
  const float* W2  = (const float*)d_in[3];
  const float* b2  = (const float*)d_in[4];
  const int*   ei  = (const int*)d_in[5];
  const int*   bat = (const int*)d_in[6];
  float* out = (float*)d_out;

  const int N  = in_sizes[0] / IN_F;     // 100000
  const int E  = in_sizes[5] / 2;        // 1600000
  const int Bg = out_size / 256;         // 128
  const int* src = ei;
  const int* dst = ei + E;

  // workspace layout (floats)
  float* ws    = (float*)d_ws;
  const long NF = (long)N * 128;
  float* bufA   = ws;                    // h1 -> c1out -> s2
  float* bufB   = bufA + NF;             // s1 -> h2pre
  float* deg    = bufB + NF;             // N
  float* dinv   = deg + N;               // N
  float* bpart  = dinv + N;              // Bg*128
  float* rootc1 = bpart + (size_t)Bg * 128;  // Bg*128
  int*   rootix = (int*)(rootc1 + (size_t)Bg * 128);

  const int T = 256;
  auto nb = [](long n, int t) { return (int)((n + t - 1) / t); };

  // zero accumulators (workspace/out are poisoned, not re-zeroed by harness)
  zero_kernel<<<nb(NF, T), T, 0, stream>>>(bufB, NF);
  zero_kernel<<<nb(N, T), T, 0, stream>>>(deg, N);
  zero_kernel<<<nb((long)Bg * 256, T), T, 0, stream>>>(out, (long)Bg * 256);

  // degrees, roots, normalization
  deg_kernel<<<nb(E, T), T, 0, stream>>>(dst, deg, E);
  root_kernel<<<nb(N, T), T, 0, stream>>>(bat, rootix, N);
  dinv_kernel<<<nb(N, T), T, 0, stream>>>(deg, dinv, N);

  // conv1 GEMM: h1 = x @ W1   [N x 128]
  gemm16_wmma<0><<<nb(N, 16), T, 0, stream>>>(x, W1, bufA, N, IN_F, nullptr, nullptr);

  // conv1 edge scatter: s1[dst] += h1[src]*dinv[src]
  edge_agg_kernel<<<nb((long)E * 32, T), T, 0, stream>>>(src, dst, dinv, bufA, bufB, E);

  // conv1 finalize in place: c1out = dinv*(s1 + h1*dinv) + b1
  finalize_kernel<<<nb(NF, T), T, 0, stream>>>(bufA, bufB, dinv, b1, NF);

  // per-graph root terms
  root_gather_kernel<<<Bg, 128, 0, stream>>>(bufA, rootix, rootc1);
  root_part_kernel<<<Bg, 128, 0, stream>>>(x, W2, rootix, bpart);

  // conv2 GEMM: h2pre = relu(c1out) @ W2[0:128] + bpart[batch]   [N x 128]
  gemm16_wmma<1><<<nb(N, 16), T, 0, stream>>>(bufA, W2, bufB, N, HID_F, bpart, bat);

  // conv2 edge scatter into zeroed bufA
  zero_kernel<<<nb(NF, T), T, 0, stream>>>(bufA, NF);
  edge_agg_kernel<<<nb((long)E * 32, T), T, 0, stream>>>(src, dst, dinv, bufB, bufA, E);

  // conv2 finalize + relu + per-graph mean-sum into out[:,0:128]
  finalize2_pool_kernel<<<nb(N, 64), T, 0, stream>>>(bufB, bufA, dinv, b2, bat, out, N);

  // divide by counts; write out[:,128:256] = conv1out at roots
  final_out_kernel<<<Bg, 256, 0, stream>>>(out, rootc1, rootix, N, Bg);
}